// GATModel_48241072668736
// MI455X (gfx1250) — compile-verified
//
#include <hip/hip_runtime.h>
#include <math.h>

#define NNODES 32768
#define NEDGES 262144
#define NGRAPH 32
#define NPG    1024
#define KTOP   820
#define HC     512
#define NEGSL  0.2f

typedef __attribute__((ext_vector_type(2))) float v2f;
typedef __attribute__((ext_vector_type(4))) float v4f;
typedef __attribute__((ext_vector_type(8))) float v8f;

#define HAVE_WMMA_F32 __has_builtin(__builtin_amdgcn_wmma_f32_16x16x4_f32)

// ---------------------------------------------------------------------------
// K0: init accumulators (ws is poisoned 0xAA once; we must re-zero every call)
// ---------------------------------------------------------------------------
__global__ __launch_bounds__(256) void init_kernel(float* __restrict__ outb,
                                                   float* __restrict__ amax,
                                                   float* __restrict__ denom) {
  int i = blockIdx.x * 256 + threadIdx.x;
  if (i < NNODES * HC) outb[i] = 0.f;
  if (i < NNODES * 8) { amax[i] = -INFINITY; denom[i] = 0.f; }
}

// ---------------------------------------------------------------------------
// K1: Out[M,512] = X[M,256] @ W[256,512] + b   (f32 WMMA 16x16x4)
// grid = (M/128, 4), block = 256 (8 waves). Wave w -> 16-row tile, 128 cols.
// ---------------------------------------------------------------------------
__global__ __launch_bounds__(256) void gemm_xw_kernel(const float* __restrict__ X,
                                                      const float* __restrict__ W,
                                                      const float* __restrict__ bvec,
                                                      float* __restrict__ Out) {
  const int wave = threadIdx.x >> 5;
  const int lane = threadIdx.x & 31;
  const int half = lane >> 4;       // 0: lanes 0-15, 1: lanes 16-31
  const int l16  = lane & 15;
  const int mtile = blockIdx.x * 8 + wave;     // 16-row tile index
  const int colbase = blockIdx.y * 128;
  const int rowA = mtile * 16 + l16;           // A-frag row (M = lane&15)

#if HAVE_WMMA_F32
  v8f acc[8];
#pragma unroll
  for (int t = 0; t < 8; ++t) acc[t] = (v8f){0.f,0.f,0.f,0.f,0.f,0.f,0.f,0.f};

  for (int k0 = 0; k0 < 256; k0 += 4) {
    // A 16x4 f32 frag: VGPR0 = K(half*2), VGPR1 = K(half*2+1), M = lane&15
    const int kb = k0 + half * 2;
    v2f a;
    a.x = X[rowA * 256 + kb];
    a.y = X[rowA * 256 + kb + 1];
#pragma unroll
    for (int t = 0; t < 8; ++t) {
      // B 4x16 f32 frag: VGPR0 = K(half*2) @ N=(lane&15), VGPR1 = K(half*2+1)
      const int col = colbase + t * 16 + l16;
      v2f b;
      b.x = W[kb * 512 + col];
      b.y = W[(kb + 1) * 512 + col];
      acc[t] = __builtin_amdgcn_wmma_f32_16x16x4_f32(
          false, a, false, b, (short)0, acc[t], false, false);
    }
  }
  // C/D layout: vgpr r, lane l -> (M = r + (l>>4)*8, N = l&15)
#pragma unroll
  for (int t = 0; t < 8; ++t) {
    const int col = colbase + t * 16 + l16;
    const float bc = bvec[col];
#pragma unroll
    for (int r = 0; r < 8; ++r) {
      const int row = mtile * 16 + r + half * 8;
      Out[row * 512 + col] = acc[t][r] + bc;
    }
  }
#else
  // VALU fallback (no WMMA) -- each lane owns 4 columns (stride 32), 16 rows.
  for (int r = 0; r < 16; ++r) {
    const int row = mtile * 16 + r;
    float s0 = 0.f, s1 = 0.f, s2 = 0.f, s3 = 0.f;
    for (int k = 0; k < 256; ++k) {
      const float xv = X[row * 256 + k];
      const float* wr = W + k * 512 + colbase + lane;
      s0 = fmaf(xv, wr[0],  s0);
      s1 = fmaf(xv, wr[32], s1);
      s2 = fmaf(xv, wr[64], s2);
      s3 = fmaf(xv, wr[96], s3);
    }
    float* orow = Out + row * 512 + colbase + lane;
    orow[0]  = s0 + bvec[colbase + lane];
    orow[32] = s1 + bvec[colbase + lane + 32];
    orow[64] = s2 + bvec[colbase + lane + 64];
    orow[96] = s3 + bvec[colbase + lane + 96];
  }
#endif
}

// ---------------------------------------------------------------------------
// float atomic max via signed/unsigned integer trick (works with -inf init)
// ---------------------------------------------------------------------------
__device__ __forceinline__ void atomicMaxFloat(float* addr, float v) {
  if (v >= 0.f) atomicMax((int*)addr, __float_as_int(v));
  else          atomicMin((unsigned int*)addr, __float_as_uint(v));
}

// ---------------------------------------------------------------------------
// K2: fused edge kernel (all b128 data paths). We (32x512, 64KB) in LDS.
// Channel map per lane: c = 128*jj + 4*lane + j  (jj,j in [0,4))
//   head(c) = c>>6 = 2*jj + (lane>>4)
// Per edge: e = ea@We via lane-broadcast, m = lrelu(xl[src]+xr[dst]+e),
// alpha_raw[h] = <m, att>, atomic-max into amax[dst,h].
// block = 256 (8 waves), EPB=128 edges/block (16 per wave).
// ---------------------------------------------------------------------------
#define EPB 128
__global__ __launch_bounds__(256) void edge_alpha_kernel(
    const float* __restrict__ xl, const float* __restrict__ xr,
    const float* __restrict__ edge_attr, const float* __restrict__ We,
    const float* __restrict__ att, const int* __restrict__ src,
    const int* __restrict__ dst, float* __restrict__ alpha,
    float* __restrict__ amax) {
  __shared__ v4f We_s4[32 * 128];   // exactly 64 KB
  {
    const v4f* Wg = (const v4f*)We;
    for (int i = threadIdx.x; i < 32 * 128; i += 256) We_s4[i] = Wg[i];
  }
  __syncthreads();

  const int wave = threadIdx.x >> 5;
  const int lane = threadIdx.x & 31;
  const int hi   = lane >> 4;       // half: selects odd/even heads

  // att values for this lane's channels: att4[jj][j] = att[128*jj + 4*lane + j]
  v4f att4[4];
#pragma unroll
  for (int jj = 0; jj < 4; ++jj) att4[jj] = ((const v4f*)att)[32 * jj + lane];

  const int e0   = blockIdx.x * EPB;
  const int eEnd = (e0 + EPB < NEDGES) ? e0 + EPB : NEDGES;
  for (int e = e0 + wave; e < eEnd; e += 8) {
    const int s = src[e];
    const int d = dst[e];
    const float ea = edge_attr[e * 32 + lane];

    // prefetch next edge's gather rows (1 cacheline per lane covers 2KB row)
    const int e2 = e + 8;
    if (e2 < eEnd) {
      const int s2 = src[e2];
      const int d2 = dst[e2];
      __builtin_prefetch(xl + s2 * 512 + lane * 16, 0, 0);
      __builtin_prefetch(xr + d2 * 512 + lane * 16, 0, 0);
    }

    // e-row: broadcast ea[k] via lane shuffles, b128 LDS reads of We
    v4f ev[4];
#pragma unroll
    for (int jj = 0; jj < 4; ++jj) ev[jj] = (v4f){0.f, 0.f, 0.f, 0.f};
#pragma unroll
    for (int k = 0; k < 32; ++k) {
      const float eak = __shfl(ea, k, 32);
      const v4f* wrow = We_s4 + k * 128 + lane;
#pragma unroll
      for (int jj = 0; jj < 4; ++jj) {
        const v4f w4 = wrow[32 * jj];
#pragma unroll
        for (int j = 0; j < 4; ++j) ev[jj][j] = fmaf(eak, w4[j], ev[jj][j]);
      }
    }

    // message + per-head partial dot with att
    float p2[4];
#pragma unroll
    for (int jj = 0; jj < 4; ++jj) p2[jj] = 0.f;
    const v4f* xls = (const v4f*)(xl + s * 512);
    const v4f* xrd = (const v4f*)(xr + d * 512);
#pragma unroll
    for (int jj = 0; jj < 4; ++jj) {
      const v4f a4 = xls[32 * jj + lane];
      const v4f b4 = xrd[32 * jj + lane];
#pragma unroll
      for (int j = 0; j < 4; ++j) {
        float mv = a4[j] + b4[j] + ev[jj][j];
        mv = (mv > 0.f) ? mv : NEGSL * mv;
        p2[jj] = fmaf(mv, att4[jj][j], p2[jj]);
      }
    }
    // reduce within each 16-lane half (heads are disjoint across halves)
#pragma unroll
    for (int off = 8; off > 0; off >>= 1) {
#pragma unroll
      for (int jj = 0; jj < 4; ++jj) p2[jj] += __shfl_xor(p2[jj], off, 32);
    }
    // lane 0 holds heads {0,2,4,6}, lane 16 holds heads {1,3,5,7}
    if ((lane & 15) == 0) {
#pragma unroll
      for (int jj = 0; jj < 4; ++jj) {
        const int h = 2 * jj + hi;
        alpha[e * 8 + h] = p2[jj];
        atomicMaxFloat(&amax[d * 8 + h], p2[jj]);
      }
    }
  }
}

// ---------------------------------------------------------------------------
// K3: ex = exp(alpha - amax[dst]); alpha <- ex; denom[dst,h] += ex
// ---------------------------------------------------------------------------
__global__ __launch_bounds__(256) void exp_denom_kernel(
    float* __restrict__ alpha, const float* __restrict__ amax,
    float* __restrict__ denom, const int* __restrict__ dst) {
  const int i = blockIdx.x * 256 + threadIdx.x;
  if (i >= NEDGES * 8) return;
  const int e = i >> 3;
  const int h = i & 7;
  const int d = dst[e];
  const float ex = expf(alpha[i] - amax[d * 8 + h]);
  alpha[i] = ex;
  atomicAdd(&denom[d * 8 + h], ex);
}

// ---------------------------------------------------------------------------
// K4: out[dst] += (ex/denom[dst]) * xl[src]   (one wave per edge, b128 loads)
// ---------------------------------------------------------------------------
__global__ __launch_bounds__(256) void scatter_kernel(
    const float* __restrict__ xl, const float* __restrict__ alpha,
    const float* __restrict__ denom, const int* __restrict__ src,
    const int* __restrict__ dst, float* __restrict__ outb) {
  const int wave = threadIdx.x >> 5;
  const int lane = threadIdx.x & 31;
  const int hi   = lane >> 4;
  const int e = blockIdx.x * 8 + wave;
  if (e >= NEDGES) return;
  const int s = src[e];
  const int d = dst[e];
  float wsel[4];
#pragma unroll
  for (int jj = 0; jj < 4; ++jj) {
    const int h = 2 * jj + hi;
    wsel[jj] = alpha[e * 8 + h] / denom[d * 8 + h];
  }
  const v4f* xls = (const v4f*)(xl + s * 512);
  float* od = outb + d * 512;
#pragma unroll
  for (int jj = 0; jj < 4; ++jj) {
    const v4f xv = xls[32 * jj + lane];
    const int cb = 128 * jj + 4 * lane;
#pragma unroll
    for (int j = 0; j < 4; ++j)
      atomicAdd(&od[cb + j], wsel[jj] * xv[j]);
  }
}

// ---------------------------------------------------------------------------
// K5: h = relu(out + bias) (in place); score[n] = tanh(h.w / ||w||)
// one wave per node, b128 row access
// ---------------------------------------------------------------------------
__global__ __launch_bounds__(256) void relu_score_kernel(
    float* __restrict__ outb, const float* __restrict__ bias,
    const float* __restrict__ pool_w, float* __restrict__ score) {
  const int wave = threadIdx.x >> 5;
  const int lane = threadIdx.x & 31;
  const int n = blockIdx.x * 8 + wave;
  float dot = 0.f, nrm = 0.f;
  v4f* row = (v4f*)(outb + n * 512);
  const v4f* b4 = (const v4f*)bias;
  const v4f* p4 = (const v4f*)pool_w;
#pragma unroll
  for (int jj = 0; jj < 4; ++jj) {
    v4f hv = row[32 * jj + lane];
    const v4f bb = b4[32 * jj + lane];
    const v4f pw = p4[32 * jj + lane];
#pragma unroll
    for (int j = 0; j < 4; ++j) {
      float v = hv[j] + bb[j];
      v = (v > 0.f) ? v : 0.f;
      hv[j] = v;
      dot = fmaf(v, pw[j], dot);
      nrm = fmaf(pw[j], pw[j], nrm);
    }
    row[32 * jj + lane] = hv;
  }
#pragma unroll
  for (int off = 16; off > 0; off >>= 1) {
    dot += __shfl_xor(dot, off, 32);
    nrm += __shfl_xor(nrm, off, 32);
  }
  if (lane == 0) score[n] = tanhf(dot / sqrtf(nrm));
}

// ---------------------------------------------------------------------------
// K6: per-graph exact top-820 by rank counting + scaled max/mean pool.
// grid = 32 graphs, block = 512 threads (one per channel).
// ---------------------------------------------------------------------------
__global__ __launch_bounds__(512) void topk_pool_kernel(
    const float* __restrict__ h, const float* __restrict__ score,
    float* __restrict__ out) {
  __shared__ float s_sc[NPG];
  __shared__ unsigned char s_sel[NPG];
  const int g = blockIdx.x;
  const float* sg = score + g * NPG;
  for (int i = threadIdx.x; i < NPG; i += 512) s_sc[i] = sg[i];
  __syncthreads();
  for (int i = threadIdx.x; i < NPG; i += 512) {
    const float si = s_sc[i];
    int rank = 0;
    for (int j = 0; j < NPG; ++j) {
      const float sj = s_sc[j];
      rank += (sj > si) || (sj == si && j < i);   // stable tie-break like top_k
    }
    s_sel[i] = (rank < KTOP) ? 1 : 0;
  }
  __syncthreads();
  const int c = threadIdx.x;
  float mx = -INFINITY, sm = 0.f;
  const float* hg = h + (size_t)g * NPG * 512;
  for (int i = 0; i < NPG; ++i) {
    if (s_sel[i]) {                                // uniform branch per block
      const float v = hg[i * 512 + c] * s_sc[i];
      mx = fmaxf(mx, v);
      sm += v;
    }
  }
  out[g * 1024 + c]       = mx;                    // global max pool
  out[g * 1024 + 512 + c] = sm * (1.0f / (float)KTOP);  // global mean pool
}

// ---------------------------------------------------------------------------
extern "C" void kernel_launch(void* const* d_in, const int* in_sizes, int n_in,
                              void* d_out, int out_size, void* d_ws, size_t ws_size,
                              hipStream_t stream) {
  const float* x         = (const float*)d_in[0];
  const float* edge_attr = (const float*)d_in[1];
  const float* Wl        = (const float*)d_in[2];
  const float* bl        = (const float*)d_in[3];
  const float* Wr        = (const float*)d_in[4];
  const float* br        = (const float*)d_in[5];
  const float* We        = (const float*)d_in[6];
  const float* att       = (const float*)d_in[7];
  const float* bias      = (const float*)d_in[8];
  const float* pool_w    = (const float*)d_in[9];
  const int*   eidx      = (const int*)d_in[10];
  const int* src = eidx;
  const int* dst = eidx + NEDGES;
  float* out = (float*)d_out;

  // workspace layout (floats)
  float* xl    = (float*)d_ws;                 // N*512
  float* xr    = xl    + (size_t)NNODES * HC;  // N*512
  float* outb  = xr    + (size_t)NNODES * HC;  // N*512 (GAT out, then h in place)
  float* alpha = outb  + (size_t)NNODES * HC;  // E*8
  float* amax  = alpha + (size_t)NEDGES * 8;   // N*8
  float* denom = amax  + (size_t)NNODES * 8;   // N*8
  float* score = denom + (size_t)NNODES * 8;   // N

  // K0: init
  init_kernel<<<(NNODES * HC + 255) / 256, 256, 0, stream>>>(outb, amax, denom);

  // K1/K2: xl = x@Wl+bl ; xr = x@Wr+br
  dim3 ggrid(NNODES / 128, 4);
  gemm_xw_kernel<<<ggrid, 256, 0, stream>>>(x, Wl, bl, xl);
  gemm_xw_kernel<<<ggrid, 256, 0, stream>>>(x, Wr, br, xr);

  // K3: fused edge attention logits + segment max
  edge_alpha_kernel<<<NEDGES / EPB, 256, 0, stream>>>(xl, xr, edge_attr, We, att,
                                                      src, dst, alpha, amax);

  // K4: exp + segment sum
  exp_denom_kernel<<<(NEDGES * 8 + 255) / 256, 256, 0, stream>>>(alpha, amax, denom, dst);

  // K5: weighted scatter-add
  scatter_kernel<<<NEDGES / 8, 256, 0, stream>>>(xl, alpha, denom, src, dst, outb);

  // K6: relu + bias + node scores
  relu_score_kernel<<<NNODES / 8, 256, 0, stream>>>(outb, bias, pool_w, score);

  // K7: top-k pooling + global max/mean pool
  topk_pool_kernel<<<NGRAPH, 512, 0, stream>>>(outb, score, out);
}